// WeightedSumGatingFuser_38285338476861
// MI455X (gfx1250) — compile-verified
//
#include <hip/hip_runtime.h>
#include <hip/hip_bf16.h>

typedef __attribute__((ext_vector_type(2))) float v2f;
typedef __attribute__((ext_vector_type(8))) float v8f;

// Types matching the async-to-LDS builtin signatures (from hipcc diagnostics):
//   b32 : (addrspace(1) int*, addrspace(3) int*, imm offset, imm cpol)
//   b128: (addrspace(1) v4i*, addrspace(3) v4i*, imm offset, imm cpol)
typedef int v4i __attribute__((vector_size(16)));
typedef __attribute__((address_space(1))) int  g_i32;
typedef __attribute__((address_space(3))) int  l_i32;
typedef __attribute__((address_space(1))) v4i  g_v4i;
typedef __attribute__((address_space(3))) v4i  l_v4i;

constexpr int Bc    = 32;
constexpr int Sc    = 1024;
constexpr int D2Dc  = 1024;
constexpr int DOUTc = 1024;
constexpr int DTEXTc= 768;
constexpr int Zc    = 256;

// ---------------------------------------------------------------------------
// gfx1250 async global->LDS copies (ASYNCcnt-tracked, bypass VGPRs).
// Guarded with __has_builtin: falls back to register staging if absent.
// ---------------------------------------------------------------------------
#define HAS_ASYNC_LDS __has_builtin(__builtin_amdgcn_global_load_async_to_lds_b32)
#define HAS_ASYNC_LDS128 __has_builtin(__builtin_amdgcn_global_load_async_to_lds_b128)

__device__ __forceinline__ void copy_g2l_b32(const float* g, float* l) {
#if HAS_ASYNC_LDS
  __builtin_amdgcn_global_load_async_to_lds_b32((g_i32*)g, (l_i32*)l, 0, 0);
#else
  *l = *g;
#endif
}

__device__ __forceinline__ void copy_g2l_b128(const float* g, float* l) {
#if HAS_ASYNC_LDS128
  __builtin_amdgcn_global_load_async_to_lds_b128((g_v4i*)g, (l_v4i*)l, 0, 0);
#else
  *reinterpret_cast<float4*>(l) = *reinterpret_cast<const float4*>(g);
#endif
}

__device__ __forceinline__ void wait_async_zero() {
#if __has_builtin(__builtin_amdgcn_s_wait_asynccnt)
  __builtin_amdgcn_s_wait_asynccnt(0);
#elif HAS_ASYNC_LDS
  asm volatile("s_wait_asynccnt 0" ::: "memory");
#endif
}

// ---------------------------------------------------------------------------
// Tiled fp32 GEMM with V_WMMA_F32_16X16X4_F32, double-buffered LDS staged by
// async global->LDS loads.
//   C[m][n] = sum_k A[m][k] * B[k][n] (+ bias)
//   ATRANS: A stored K-major in memory (A[m][k] = Aglob[k*lda + m])
//   BIAS: 0 = none, 1 = per-column bias[n], 2 = per-row bias[m]
// Workgroup: 256 threads (8 waves). Block tile 128(M) x 64(N), K-step 16.
// Wave w computes rows [w*16, w*16+16) x 64 columns -> 4 v8f accumulators.
// LDS: As[2][128][20] row-major (K contiguous, 80B rows -> B128-alignable and
// conflict-free), Bs[2][64][20] = B tile transposed [n][k] so fragment
// fetches are consecutive-K b64 loads.
// Per iteration: wait own ASYNCcnt==0 -> barrier (buffer `cur` ready for all)
// -> issue async prefetch of `cur^1` -> 16 WMMAs from `cur`. Reads of `cur`
// are complete before the wave reaches the next barrier (WMMA forces
// s_wait_dscnt), so the following prefetch into `cur` can never race them.
// ---------------------------------------------------------------------------
template <bool ATRANS, int BIAS>
__launch_bounds__(256)
__global__ void gemm128x64_wmma(const float* __restrict__ A,
                                const float* __restrict__ Bm,
                                const float* __restrict__ bias,
                                float* __restrict__ C,
                                int lda, int ldb, int ldc, int K,
                                long long sA, long long sB, long long sC,
                                long long sBias) {
  const int bn    = blockIdx.x * 64;
  const int bm    = blockIdx.y * 128;
  const int batch = blockIdx.z;
  A  += (long long)batch * sA;
  Bm += (long long)batch * sB;
  C  += (long long)batch * sC;
  if (BIAS != 0) bias += (long long)batch * sBias;

  __shared__ float As[2][128 * 20];
  __shared__ float Bs[2][64 * 20];

  const int tid     = threadIdx.x;
  const int lane    = tid & 31;
  const int wave    = tid >> 5;
  const int halfsel = lane >> 4;  // 0: lanes 0-15, 1: lanes 16-31
  const int l16     = lane & 15;

  v8f acc0 = {}, acc1 = {}, acc2 = {}, acc3 = {};

  // ---- stage one 128x16 A tile + 16x64 B tile into LDS buffer `buf` ----
  auto stage = [&](int buf, int kt) {
    if (!ATRANS) {
      // A row-major [M][K]: B128 per slot (4 floats along K), 16B-aligned.
#pragma unroll
      for (int s = 0; s < 2; ++s) {
        const int slot = tid + s * 256;     // 512 B128 slots
        const int row  = slot >> 2;         // 0..127
        const int kq   = (slot & 3) << 2;   // 0,4,8,12
        copy_g2l_b128(A + (long long)(bm + row) * lda + kt + kq,
                      &As[buf][row * 20 + kq]);
      }
    } else {
      // A K-major [K][M]: per-lane B32 with on-the-fly transpose; lanes read
      // consecutive m -> coalesced 128B per wave.
#pragma unroll
      for (int s = 0; s < 8; ++s) {
        const int slot = tid + s * 256;     // 2048 B32 slots
        const int krow = slot >> 7;         // 0..15
        const int m    = slot & 127;        // 0..127
        copy_g2l_b32(A + (long long)(kt + krow) * lda + bm + m,
                     &As[buf][m * 20 + krow]);
      }
    }
    // B [K][N] -> Bs[n][k] transposed, per-lane B32 (consecutive n per lane).
#pragma unroll
    for (int s = 0; s < 4; ++s) {
      const int slot = tid + s * 256;       // 1024 B32 slots
      const int krow = slot >> 6;           // 0..15
      const int n    = slot & 63;           // 0..63
      copy_g2l_b32(Bm + (long long)(kt + krow) * ldb + bn + n,
                   &Bs[buf][n * 20 + krow]);
    }
  };

  stage(0, 0);
  const int nsteps = K >> 4;
  for (int step = 0; step < nsteps; ++step) {
    const int cur = step & 1;
    wait_async_zero();
    __syncthreads();
    if (step + 1 < nsteps) stage(cur ^ 1, (step + 1) << 4);

    // Fragment bases. A frag (16x4 f32): row M = lane%16, VGPR0/1 hold
    // K = k0 + 2*halfsel + {0,1}. B frag (4x16): col N = lane%16, same K
    // selection; Bs is [n][k] so both are consecutive-K b64 loads.
    const float* aBase = &As[cur][(wave * 16 + l16) * 20 + (halfsel << 1)];
    const float* bBase = &Bs[cur][l16 * 20 + (halfsel << 1)];

#pragma unroll
    for (int ks = 0; ks < 4; ++ks) {
      const int k0 = ks << 2;
      const v2f a  = *(const v2f*)(aBase + k0);
      const v2f b0 = *(const v2f*)(bBase + 0 * 16 * 20 + k0);
      const v2f b1 = *(const v2f*)(bBase + 1 * 16 * 20 + k0);
      const v2f b2 = *(const v2f*)(bBase + 2 * 16 * 20 + k0);
      const v2f b3 = *(const v2f*)(bBase + 3 * 16 * 20 + k0);
      acc0 = __builtin_amdgcn_wmma_f32_16x16x4_f32(false, a, false, b0,
                                                   (short)0, acc0, false, false);
      acc1 = __builtin_amdgcn_wmma_f32_16x16x4_f32(false, a, false, b1,
                                                   (short)0, acc1, false, false);
      acc2 = __builtin_amdgcn_wmma_f32_16x16x4_f32(false, a, false, b2,
                                                   (short)0, acc2, false, false);
      acc3 = __builtin_amdgcn_wmma_f32_16x16x4_f32(false, a, false, b3,
                                                   (short)0, acc3, false, false);
    }
  }

  // ---- epilogue: C/D layout VGPR r -> M = r + 8*halfsel, N = lane&15 ----
  const int row0 = bm + wave * 16 + halfsel * 8;
  const int col0 = bn + l16;
  v8f accs[4] = {acc0, acc1, acc2, acc3};
#pragma unroll
  for (int nt = 0; nt < 4; ++nt) {
    const int col = col0 + nt * 16;
    const float cb = (BIAS == 1) ? bias[col] : 0.0f;
#pragma unroll
    for (int r = 0; r < 8; ++r) {
      float val = accs[nt][r] + cb;
      if (BIAS == 2) val += bias[row0 + r];
      C[(long long)(row0 + r) * ldc + col] = val;
    }
  }
}

// ---------------------------------------------------------------------------
// Column softmax over S: one block per (b, z); elements stride Z in memory.
// ---------------------------------------------------------------------------
__launch_bounds__(256)
__global__ void softmax_s_kernel(float* __restrict__ scores) {
  const int bz = blockIdx.x;
  const int b  = bz >> 8;       // Z = 256
  const int z  = bz & 255;
  float* col = scores + (long long)b * (Sc * Zc) + z;

  __shared__ float red[256];
  const int t = threadIdx.x;

  float v[4];
  float m = -3.402823466e+38f;
#pragma unroll
  for (int i = 0; i < 4; ++i) {
    v[i] = col[(long long)(t + i * 256) * Zc];
    m = fmaxf(m, v[i]);
  }
  red[t] = m;
  __syncthreads();
  for (int off = 128; off > 0; off >>= 1) {
    if (t < off) red[t] = fmaxf(red[t], red[t + off]);
    __syncthreads();
  }
  m = red[0];
  __syncthreads();

  float s = 0.0f;
#pragma unroll
  for (int i = 0; i < 4; ++i) {
    v[i] = __expf(v[i] - m);
    s += v[i];
  }
  red[t] = s;
  __syncthreads();
  for (int off = 128; off > 0; off >>= 1) {
    if (t < off) red[t] += red[t + off];
    __syncthreads();
  }
  const float inv = 1.0f / red[0];
#pragma unroll
  for (int i = 0; i < 4; ++i) {
    col[(long long)(t + i * 256) * Zc] = v[i] * inv;
  }
}

// ---------------------------------------------------------------------------
extern "C" void kernel_launch(void* const* d_in, const int* in_sizes, int n_in,
                              void* d_out, int out_size, void* d_ws,
                              size_t ws_size, hipStream_t stream) {
  const float* feat_2d   = (const float*)d_in[0];  // (B,S,D2D)
  const float* feat_3d   = (const float*)d_in[1];  // (B,1,DOUT)
  const float* feat_text = (const float*)d_in[2];  // (B,DTEXT,Z)
  const float* W2d       = (const float*)d_in[3];  // (D2D,DOUT)
  const float* b2d       = (const float*)d_in[4];  // (DOUT)
  const float* Wtext     = (const float*)d_in[5];  // (DTEXT,DOUT)
  const float* btext     = (const float*)d_in[6];  // (DOUT)
  float* out = (float*)d_out;                      // (B,DOUT,Z)

  float* ws     = (float*)d_ws;
  float* f2d    = ws;                                      // B*S*DOUT
  float* ftext  = f2d + (long long)Bc * Sc * DOUTc;        // B*DOUT*Z
  float* scores = ftext + (long long)Bc * DOUTc * Zc;      // B*S*Z (in-place softmax)

  const dim3 blk(256);

  // 1) f2d = feat_2d @ W2d + b2d      : (B*S x D2D)(D2D x DOUT), col bias
  gemm128x64_wmma<false, 1><<<dim3(DOUTc / 64, (Bc * Sc) / 128, 1), blk, 0, stream>>>(
      feat_2d, W2d, b2d, f2d, D2Dc, DOUTc, DOUTc, D2Dc, 0, 0, 0, 0);

  // 2) ftext[b] = Wtext^T @ feat_text[b] + btext : (DOUT x DTEXT)(DTEXT x Z), row bias
  gemm128x64_wmma<true, 2><<<dim3(Zc / 64, DOUTc / 128, Bc), blk, 0, stream>>>(
      Wtext, feat_text, btext, ftext, DOUTc, Zc, Zc, DTEXTc,
      0, (long long)DTEXTc * Zc, (long long)DOUTc * Zc, 0);

  // 3) scores[b] = f2d[b] @ ftext[b]  : (S x DOUT)(DOUT x Z)
  gemm128x64_wmma<false, 0><<<dim3(Zc / 64, Sc / 128, Bc), blk, 0, stream>>>(
      f2d, ftext, nullptr, scores, DOUTc, Zc, Zc, DOUTc,
      (long long)Sc * DOUTc, (long long)DOUTc * Zc, (long long)Sc * Zc, 0);

  // 4) softmax over S, in place -> weights
  softmax_s_kernel<<<Bc * Zc, blk, 0, stream>>>(scores);

  // 5) out[b] = f2d[b]^T @ weights[b] + feat_3d[b] : (DOUT x S)(S x Z), row bias
  gemm128x64_wmma<true, 2><<<dim3(Zc / 64, DOUTc / 128, Bc), blk, 0, stream>>>(
      f2d, scores, feat_3d, out, DOUTc, Zc, Zc, Sc,
      (long long)Sc * DOUTc, (long long)Sc * Zc, (long long)DOUTc * Zc, DOUTc);
}